// NUFFTLayerMultiChannelInitMixed_15393162789281
// MI455X (gfx1250) — compile-verified
//
#include <hip/hip_runtime.h>
#include <math.h>

// M = 2001 grid points, L = 2*pi, tau = 12/M^2.
// e_ch[b,n] = (sum_j u_j)^T A_ch u_n - u_n^T A_ch u_n, A_ch circulant with row c_ch.
// Pipeline: k1 build c_ch[d] ; k2 spread S_b (LDS atomics) ; k3 V = C*S via
// V_WMMA_F32_16X16X4_F32 ; k4 per-particle energies.

#define MPTS   2001
#define HALFK  1000          // p = 1..1000 (symmetric spectrum), KGRID = integers
#define KW     33            // Gaussian support window (fp32-exact beyond ~12)
#define HALFW  16
#define NPART  1024
#define NBATCH 16
#define KPAD   2004          // 2001 rounded up to multiple of 4 (zero padded)

typedef float v2f __attribute__((ext_vector_type(2)));
typedef float v8f __attribute__((ext_vector_type(8)));

__device__ __forceinline__ float tau_f()     { return (float)(12.0 / (2001.0 * 2001.0)); }
__device__ __forceinline__ float inv4tau_f() { return 1.0f / (4.0f * tau_f()); }

// --------------------------------------------------------------------------
// Kernel 1: c_ch[d] = (scale/M) * [ w[0] + 2 * sum_{p=1}^{1000} w[p] cos(2 pi p d / M) ]
// w_ch[p] = mult_ch(p) * deconv(p)^2 ; KGRID is exactly the integers -1000..1000.
// --------------------------------------------------------------------------
__global__ void k_build_c(const float* __restrict__ shift0, const float* __restrict__ shift1,
                          const float* __restrict__ amp0,  const float* __restrict__ amp1,
                          float* __restrict__ cOut /* [2][MPTS] */) {
    __shared__ float wb[HALFK + 1];
    const int ch = blockIdx.y;
    const float shift = (ch == 0) ? shift0[0] : shift1[0];
    const float mu    = (ch == 0) ? 1.0f : 0.5f;
    const float amp   = (ch == 0) ? amp0[0] : amp1[0];
    const float mus   = mu * shift;
    const float mus2  = mus * mus;

    const float TAU   = tau_f();
    const float sqc   = sqrtf((float)M_PI / TAU);            // sqrt(pi/tau)
    const float fourpi = 4.0f * (float)M_PI;
    const float pref  = (float)(1.0 / (2.0 * M_PI * 2001.0 * 2001.0)); // scale/M

    for (int p = threadIdx.x; p <= HALFK; p += blockDim.x) {
        float pf   = (float)p;
        float dc   = sqc * __expf(pf * pf * TAU);            // deconv(p)
        float mult = amp * fourpi / (pf * pf + mus2);
        float w    = pref * mult * dc * dc;
        wb[p] = (p == 0) ? w : 2.0f * w;                     // fold symmetry factor
    }
    __syncthreads();

    const int d = blockIdx.x * blockDim.x + threadIdx.x;
    if (d >= MPTS) return;
    const float th = (float)(2.0 * M_PI / 2001.0);
    float acc = wb[0];
    for (int p = 1; p <= HALFK; ++p) {
        int r = (p * d) % MPTS;                              // exact angle reduction
        acc += wb[p] * __cosf((float)r * th);
    }
    cOut[ch * MPTS + d] = acc;
}

// --------------------------------------------------------------------------
// Kernel 2: spread S_b[m] = sum_n g_n[m] via LDS atomics; store k-major St[k][b].
// xgrid matches np.linspace: float( m * (2*pi/2001 in double) ).
// --------------------------------------------------------------------------
__global__ void k_spread(const float* __restrict__ x, float* __restrict__ St /* [KPAD][16] */) {
    __shared__ float S[MPTS];
    const int b = blockIdx.x;
    for (int i = threadIdx.x; i < MPTS; i += blockDim.x) S[i] = 0.0f;
    __syncthreads();

    const float inv4tau = inv4tau_f();
    const float MoverL  = (float)(2001.0 / (2.0 * M_PI));
    const double H      = (2.0 * M_PI) / 2001.0;

    for (int n = threadIdx.x; n < NPART; n += blockDim.x) {
        float xv = x[b * NPART + n];
        int   i0 = (int)floorf(xv * MoverL);
        #pragma unroll
        for (int i = 0; i < KW; ++i) {
            int   m    = i0 - HALFW + i;
            float xg   = (float)((double)m * H);
            float diff = xv - xg;                            // matches ref (one image term)
            float g    = __expf(-(diff * diff) * inv4tau);
            int mm = m; if (mm < 0) mm += MPTS; else if (mm >= MPTS) mm -= MPTS;
            atomicAdd(&S[mm], g);                            // ds_add_f32
        }
    }
    __syncthreads();
    for (int k = threadIdx.x; k < KPAD; k += blockDim.x)
        St[k * NBATCH + b] = (k < MPTS) ? S[k] : 0.0f;       // zero-pad K to mult of 4
}

// --------------------------------------------------------------------------
// Kernel 3: V_ch[m][b] = sum_k c_ch[(m-k) mod M] * S_b[k]  ==  16x16 D-tiles of
// C[2001x2001] x S[2001x16] via V_WMMA_F32_16X16X4_F32. 128 threads = 4 waves,
// each wave owns one 16-row tile; circulant rows generated from c in LDS.
// --------------------------------------------------------------------------
__global__ void k_conv_wmma(const float* __restrict__ c, const float* __restrict__ St,
                            float* __restrict__ Vt /* [2][KPAD][16] */) {
    __shared__ float cext[2080];                             // c[(m0b + t - 2003) mod M]
    const int ch  = blockIdx.y;
    const float* cc = c + ch * MPTS;
    const int m0b = blockIdx.x * 64;                         // 4 tiles * 16 rows

    for (int t = threadIdx.x; t < 2067; t += blockDim.x) {
        int j = m0b + t - 2003;
        while (j < 0)      j += MPTS;
        while (j >= MPTS)  j -= MPTS;
        cext[t] = cc[j];
    }
    __syncthreads();

    const int wave = threadIdx.x >> 5;                       // wave32
    const int lane = threadIdx.x & 31;
    const int m0   = m0b + wave * 16;
    if (m0 >= MPTS) return;                                  // wave-uniform: EXEC stays full

    const int rn = lane & 15;                                // A row / B,D column
    const int kh = lane >> 4;                                // K-half (ISA 16x16x4 layout)
    const int lr = wave * 16 + rn;

    v8f acc = {};
    const int   abase = lr + 2003 - 2 * kh;                  // A[row][2kh+j] = cext[abase - j - k0]
    const float* bbase = St + (2 * kh) * NBATCH + rn;        // B[2kh+j][n]   = St[(k0+2kh+j)*16+n]

    for (int k0 = 0; k0 < KPAD; k0 += 4) {
        v2f a, bv;
        a.x  = cext[abase - k0];
        a.y  = cext[abase - 1 - k0];
        bv.x = bbase[k0 * NBATCH];
        bv.y = bbase[k0 * NBATCH + NBATCH];
        acc = __builtin_amdgcn_wmma_f32_16x16x4_f32(
                  false, a, false, bv, (short)0, acc, false, false);
    }

    float* vout = Vt + ch * (KPAD * NBATCH);
    #pragma unroll
    for (int r = 0; r < 8; ++r) {                            // D: VGPR r -> rows r / r+8
        int gm = m0 + r + 8 * kh;
        if (gm < MPTS) vout[gm * NBATCH + rn] = acc[r];
    }
}

// --------------------------------------------------------------------------
// Kernel 4: per-particle  e_ch = g . V_ch  -  sum_{i,j} g_i g_j c_ch[|i-j|]
// (c[d] = c[M-d], support window => only c[0..32] needed for the diagonal).
// --------------------------------------------------------------------------
__global__ void k_energy(const float* __restrict__ x, const float* __restrict__ c,
                         const float* __restrict__ Vt, float* __restrict__ out) {
    __shared__ float cs0[KW], cs1[KW];
    for (int t = threadIdx.x; t < KW; t += blockDim.x) {
        cs0[t] = c[t];
        cs1[t] = c[MPTS + t];
    }
    __syncthreads();

    const int gid = blockIdx.x * blockDim.x + threadIdx.x;
    if (gid >= NBATCH * NPART) return;
    const int b = gid >> 10;

    const float inv4tau = inv4tau_f();
    const float MoverL  = (float)(2001.0 / (2.0 * M_PI));
    const double H      = (2.0 * M_PI) / 2001.0;

    float xv = x[gid];
    int   i0 = (int)floorf(xv * MoverL);

    float g[KW];
    int   mm[KW];
    #pragma unroll
    for (int i = 0; i < KW; ++i) {
        int   m    = i0 - HALFW + i;
        float xg   = (float)((double)m * H);
        float diff = xv - xg;
        g[i] = __expf(-(diff * diff) * inv4tau);
        if (m < 0) m += MPTS; else if (m >= MPTS) m -= MPTS;
        mm[i] = m;
    }

    const float* v0 = Vt;
    const float* v1 = Vt + KPAD * NBATCH;
    float et0 = 0.0f, et1 = 0.0f;
    #pragma unroll
    for (int i = 0; i < KW; ++i) {
        int idx = mm[i] * NBATCH + b;
        et0 = fmaf(g[i], v0[idx], et0);
        et1 = fmaf(g[i], v1[idx], et1);
    }

    float ed0 = 0.0f, ed1 = 0.0f;
    for (int t = 0; t < KW; ++t) {
        float qt = 0.0f;
        for (int i = 0; i + t < KW; ++i) qt = fmaf(g[i], g[i + t], qt);
        float f = (t == 0) ? 1.0f : 2.0f;
        ed0 = fmaf(f * cs0[t], qt, ed0);
        ed1 = fmaf(f * cs1[t], qt, ed1);
    }

    out[gid * 2 + 0] = et0 - ed0;
    out[gid * 2 + 1] = et1 - ed1;
}

// --------------------------------------------------------------------------
extern "C" void kernel_launch(void* const* d_in, const int* in_sizes, int n_in,
                              void* d_out, int out_size, void* d_ws, size_t ws_size,
                              hipStream_t stream) {
    const float* x      = (const float*)d_in[0];  // [16,1024]
    const float* shift0 = (const float*)d_in[1];
    const float* shift1 = (const float*)d_in[2];
    const float* amp0   = (const float*)d_in[3];
    const float* amp1   = (const float*)d_in[4];
    float* out = (float*)d_out;                   // [16,1024,2]

    float* W  = (float*)d_ws;
    float* c  = W;                                // [2][2001]   (pad to 4096)
    float* St = W + 4096;                         // [2004][16]  = 32064
    float* Vt = W + 4096 + 32064;                 // [2][2004][16]

    k_build_c <<<dim3(8, 2),  256, 0, stream>>>(shift0, shift1, amp0, amp1, c);
    k_spread  <<<NBATCH,      256, 0, stream>>>(x, St);
    k_conv_wmma<<<dim3(32, 2), 128, 0, stream>>>(c, St, Vt);
    k_energy  <<<(NBATCH * NPART + 255) / 256, 256, 0, stream>>>(x, c, Vt, out);
}